// Attention_42949672960850
// MI455X (gfx1250) — compile-verified
//
#include <hip/hip_runtime.h>
#include <hip/hip_bf16.h>
#include <math.h>

// ---------------------------------------------------------------------------
// Shapes (compile-time constants from the reference):
//   B=8, OUT_LEN=64, IN_LEN=512, DEC=512, ATTN=512
// ---------------------------------------------------------------------------
#define BB       8
#define OUT_LEN  64
#define IN_LEN   512
#define DEC      512
#define ATTN     512

typedef __attribute__((ext_vector_type(2))) float v2f;
typedef __attribute__((ext_vector_type(8))) float v8f;

// Explicit global-address-space pointers: guarantees global_load/global_store
// (flat_load also ticks DScnt and arbitrates against the LDS pipe).
typedef __attribute__((address_space(1))) const float* gcp;
typedef __attribute__((address_space(1)))       float* gfp;

// ---------------------------------------------------------------------------
// Inner K-loop: double-buffered fragments so loads for step k+4 are in
// flight while the WMMAs for step k execute.
//   Abase: A + row(m0+r)*LDA + half*2      (per-lane A base, k=0)
//   Wbase: W + (half*2)*LDW + n0 + r       (per-lane B base, k=0)
// Fragment layouts per CDNA5 ISA 7.12.2 (wave32):
//   A 16x4 :  M = lane%16, VGPR j -> K = (lane/16)*2 + j
//   B 4x16 :  N = lane%16, VGPR j -> K = (lane/16)*2 + j
//   C/D    :  N = lane%16, VGPR j -> M = (lane/16)*8 + j
// ---------------------------------------------------------------------------
template<int KLEN, int LDA, int LDW, int MT>
__device__ __forceinline__ void gemm_loop(gcp Abase, gcp Wbase, v8f (&acc)[MT])
{
    v2f bN, aN[MT];
    // preload k = 0
    bN.x = Wbase[0];
    bN.y = Wbase[LDW];
#pragma unroll
    for (int t = 0; t < MT; ++t) {
        aN[t].x = Abase[(size_t)t * 16 * LDA + 0];
        aN[t].y = Abase[(size_t)t * 16 * LDA + 1];
    }

#pragma unroll 2
    for (int k0 = 0; k0 < KLEN; k0 += 4) {
        // rotate: current <- next
        v2f bC = bN;
        v2f aC[MT];
#pragma unroll
        for (int t = 0; t < MT; ++t) aC[t] = aN[t];

        // issue loads for k0+4 while the WMMAs below consume k0
        if (k0 + 4 < KLEN) {
            bN.x = Wbase[(size_t)(k0 + 4) * LDW];
            bN.y = Wbase[(size_t)(k0 + 5) * LDW];
#pragma unroll
            for (int t = 0; t < MT; ++t) {
                aN[t].x = Abase[(size_t)t * 16 * LDA + k0 + 4];
                aN[t].y = Abase[(size_t)t * 16 * LDA + k0 + 5];
            }
        }

#pragma unroll
        for (int t = 0; t < MT; ++t)
            acc[t] = __builtin_amdgcn_wmma_f32_16x16x4_f32(
                         false, aC[t], false, bC, (short)0, acc[t], false, false);
    }
}

// ---------------------------------------------------------------------------
// f32 WMMA GEMM, fully constant-strided:
//   D[M,N] = act( [A0 | A1][M, K] @ W[K,N] + bias[N] )
// Concat split at compile-time KSPLIT (second phase folds away if KSPLIT==K).
// Block = 128 threads = 4 waves; wave w owns N-tile (blockIdx.y*4 + w).
// Each wave computes a 64x16 strip: MT=4 M-subtiles share one B fragment
// (4 WMMAs per 2 strided B loads + 4 contiguous 8-byte A loads; 4
// independent accumulation chains).
// ---------------------------------------------------------------------------
template<int K, int KSPLIT, int LDA0, int LDA1, int LDW, int LDD, bool TANH, int MT>
__global__ void __launch_bounds__(128)
wmma_gemm(const float* __restrict__ A0,
          const float* __restrict__ A1,
          const float* __restrict__ W,
          const float* __restrict__ bias,
          float* __restrict__ D,
          long strideA, long strideW, long strideD)
{
    const int lane = threadIdx.x & 31;
    const int wave = threadIdx.x >> 5;                 // 0..3
    const int half = lane >> 4;                        // 0 or 1
    const int r    = lane & 15;
    const int m0   = blockIdx.x * (16 * MT);
    const int n0   = (blockIdx.y * 4 + wave) * 16;
    const int bz   = blockIdx.z;

    v8f acc[MT];
#pragma unroll
    for (int t = 0; t < MT; ++t)
        acc[t] = (v8f){0.f, 0.f, 0.f, 0.f, 0.f, 0.f, 0.f, 0.f};

    gcp Wp = (gcp)(W + (size_t)bz * (size_t)strideW
                     + (size_t)(half << 1) * LDW + n0 + r);

    // phase 1: A0 over k in [0, KSPLIT)
    {
        gcp Ab = (gcp)(A0 + (size_t)bz * (size_t)strideA
                          + (size_t)(m0 + r) * LDA0 + (half << 1));
        gemm_loop<KSPLIT, LDA0, LDW, MT>(Ab, Wp, acc);
    }

    // phase 2: A1 over k in [KSPLIT, K)  (compile-time folded when empty)
    if (KSPLIT < K) {
        gcp Ab = (gcp)(A1 + (size_t)(m0 + r) * LDA1 + (half << 1));
        gemm_loop<(K - KSPLIT > 0 ? K - KSPLIT : 0), LDA1, LDW, MT>(
            Ab, Wp + (size_t)KSPLIT * LDW, acc);
    }

    // epilogue: d[j] -> (m0 + t*16 + half*8 + j, n0 + r)
    const float bv = bias ? bias[n0 + r] : 0.f;
    gfp Dbase = (gfp)(D + (size_t)bz * (size_t)strideD + n0 + r);
#pragma unroll
    for (int t = 0; t < MT; ++t) {
        gfp drow = Dbase + (size_t)(m0 + t * 16 + half * 8) * LDD;
#pragma unroll
        for (int j = 0; j < 8; ++j) {
            float v = acc[t][j] + bv;
            if (TANH) v = tanhf(v);
            drow[(size_t)j * LDD] = v;
        }
    }
}

// ---------------------------------------------------------------------------
// Fused additive-attention score + softmax (the real hot spot: 134M tanh).
//   scores[b,o,i] = sum_d tanh(MA[b,i,d] + MO[b,o,d]) * q[d] + qb
//   attn[b,o,:]   = softmax_i(scores[b,o,:])
// One 256-thread block per (b,o). MO row + q staged in LDS; each thread
// streams two MA rows (float4), then block-wide max/sum reductions.
// ---------------------------------------------------------------------------
__global__ void __launch_bounds__(256)
score_softmax_kernel(const float* __restrict__ MA,   // [B, IN, DEC]
                     const float* __restrict__ MO,   // [B, OUT, DEC]
                     const float* __restrict__ qw,   // [DEC]
                     const float* __restrict__ qb,   // [1]
                     float* __restrict__ ATT)        // [B, OUT, IN]
{
    const int bo = blockIdx.x;            // 0 .. B*OUT_LEN-1
    const int b  = bo >> 6;               // /64
    const int t  = threadIdx.x;           // 0..255

    __shared__ float s_mo[DEC];
    __shared__ float s_q[DEC];
    __shared__ float s_sc[IN_LEN];
    __shared__ float s_red[256];

    const float* __restrict__ mo = MO + (size_t)bo * DEC;
    s_mo[t]       = mo[t];
    s_mo[t + 256] = mo[t + 256];
    s_q[t]        = qw[t];
    s_q[t + 256]  = qw[t + 256];
    __syncthreads();

    const float qb0 = qb[0];
#pragma unroll
    for (int ii = 0; ii < 2; ++ii) {
        const int i = t + ii * 256;
        gcp ma = (gcp)(MA + ((size_t)b * IN_LEN + i) * DEC);
        float s = 0.f;
        for (int d = 0; d < DEC; d += 4) {
            const float x0 = ma[d + 0], x1 = ma[d + 1];
            const float x2 = ma[d + 2], x3 = ma[d + 3];
            s += tanhf(x0 + s_mo[d + 0]) * s_q[d + 0];
            s += tanhf(x1 + s_mo[d + 1]) * s_q[d + 1];
            s += tanhf(x2 + s_mo[d + 2]) * s_q[d + 2];
            s += tanhf(x3 + s_mo[d + 3]) * s_q[d + 3];
        }
        s_sc[i] = s + qb0;
    }
    __syncthreads();

    // block max over 512 scores
    s_red[t] = fmaxf(s_sc[t], s_sc[t + 256]);
    __syncthreads();
    for (int off = 128; off > 0; off >>= 1) {
        if (t < off) s_red[t] = fmaxf(s_red[t], s_red[t + off]);
        __syncthreads();
    }
    const float mx = s_red[0];
    __syncthreads();

    const float e0 = __expf(s_sc[t]       - mx);
    const float e1 = __expf(s_sc[t + 256] - mx);
    s_red[t] = e0 + e1;
    __syncthreads();
    for (int off = 128; off > 0; off >>= 1) {
        if (t < off) s_red[t] += s_red[t + off];
        __syncthreads();
    }
    const float inv = 1.f / s_red[0];

    gfp arow = (gfp)(ATT + (size_t)bo * IN_LEN);
    arow[t]       = e0 * inv;
    arow[t + 256] = e1 * inv;
}

// ---------------------------------------------------------------------------
// Launch pipeline
// ---------------------------------------------------------------------------
extern "C" void kernel_launch(void* const* d_in, const int* in_sizes, int n_in,
                              void* d_out, int out_size, void* d_ws, size_t ws_size,
                              hipStream_t stream)
{
    (void)in_sizes; (void)n_in; (void)out_size; (void)ws_size;

    const float* output  = (const float*)d_in[0];  // [B, OUT, DEC]
    const float* context = (const float*)d_in[1];  // [B, IN, ATTN]
    const float* dec_w   = (const float*)d_in[2];  // [DEC, DEC]
    const float* dec_b   = (const float*)d_in[3];  // [DEC]
    const float* attn_w  = (const float*)d_in[4];  // [ATTN, DEC]
    const float* attn_b  = (const float*)d_in[5];  // [DEC]
    const float* query_w = (const float*)d_in[6];  // [DEC]
    const float* query_b = (const float*)d_in[7];  // [1]
    const float* out_w   = (const float*)d_in[8];  // [DEC+ATTN, DEC]
    const float* out_b   = (const float*)d_in[9];  // [DEC]

    // workspace: mapped_attn (8 MB) | mapped_out (1 MB) | mix (1 MB)
    float* mapped_attn = (float*)d_ws;                              // B*IN*DEC
    float* mapped_out  = mapped_attn + (size_t)BB * IN_LEN * DEC;   // B*OUT*DEC
    float* mix         = mapped_out  + (size_t)BB * OUT_LEN * DEC;  // B*OUT*ATTN

    // d_out = [out : B*OUT*DEC][attn : B*OUT*IN]
    float* outp = (float*)d_out;
    float* attn = outp + (size_t)BB * OUT_LEN * DEC;

    // 1) mapped_attn[B*IN, DEC] = context[B*IN, ATTN] @ attn_w + attn_b
    {
        dim3 grid((BB * IN_LEN) / 64, DEC / 64, 1);
        wmma_gemm<512, 512, 512, 512, 512, 512, false, 4><<<grid, 128, 0, stream>>>(
            context, context, attn_w, attn_b, mapped_attn, 0, 0, 0);
    }

    // 2) mapped_out[B*OUT, DEC] = output[B*OUT, DEC] @ dec_w + dec_b
    {
        dim3 grid((BB * OUT_LEN) / 64, DEC / 64, 1);
        wmma_gemm<512, 512, 512, 512, 512, 512, false, 4><<<grid, 128, 0, stream>>>(
            output, output, dec_w, dec_b, mapped_out, 0, 0, 0);
    }

    // 3) fused tanh-score + softmax -> attn (second half of d_out)
    score_softmax_kernel<<<BB * OUT_LEN, 256, 0, stream>>>(
        mapped_attn, mapped_out, query_w, query_b, attn);

    // 4) mix[b][OUT, ATTN] = attn[b][OUT, IN] @ context[b][IN, ATTN]  (batched)
    {
        dim3 grid(OUT_LEN / 64, ATTN / 64, BB);
        wmma_gemm<512, 512, 512, 512, 512, 512, false, 4><<<grid, 128, 0, stream>>>(
            attn, attn, context, /*bias=*/nullptr, mix,
            /*strideA=*/(long)OUT_LEN * IN_LEN,
            /*strideW=*/(long)IN_LEN * ATTN,
            /*strideD=*/(long)OUT_LEN * ATTN);
    }

    // 5) out[B*OUT, DEC] = tanh( [mix | output] @ out_w + out_b )
    //    concat handled via compile-time K-split at ATTN (=512), K total 1024
    {
        dim3 grid((BB * OUT_LEN) / 64, DEC / 64, 1);
        wmma_gemm<1024, 512, 512, 512, 512, 512, true, 4><<<grid, 128, 0, stream>>>(
            mix, output, out_w, out_b, outp, 0, 0, 0);
    }
}